// Z_QuadraticTriDiagLayer_48017734369897
// MI455X (gfx1250) — compile-verified
//
#include <hip/hip_runtime.h>
#include <math.h>

// Problem dimensions (fixed by the reference).
#define N_ 4096
#define M_ 2048
#define B_ 64

typedef float v2f __attribute__((ext_vector_type(2)));
typedef float v8f __attribute__((ext_vector_type(8)));
typedef unsigned int v4u __attribute__((ext_vector_type(4)));
typedef int v8i __attribute__((ext_vector_type(8)));
typedef int v4i __attribute__((ext_vector_type(4)));

#if defined(__has_builtin)
#if __has_builtin(__builtin_amdgcn_tensor_load_to_lds) && \
    __has_builtin(__builtin_amdgcn_s_wait_tensorcnt)
#define HAVE_TDM 1
#endif
#endif
#ifndef HAVE_TDM
#define HAVE_TDM 0
#endif

// ---------------------------------------------------------------------------
// Eigenvalues of symmetric tridiagonal T via Sturm-sequence bisection.
// ---------------------------------------------------------------------------
__device__ __forceinline__ int sturm_count(const float* __restrict__ dg,
                                           const float* __restrict__ od,
                                           int n, float x) {
  int cnt = 0;
  float q = dg[0] - x;
  if (q < 0.0f) cnt++;
  for (int i = 1; i < n; ++i) {
    float e = od[i - 1];
    float denom = q;
    if (fabsf(denom) < 1e-30f) denom = (denom < 0.0f) ? -1e-30f : 1e-30f;
    q = (dg[i] - x) - (e * e) / denom;
    if (q < 0.0f) cnt++;
  }
  return cnt;
}

__global__ __launch_bounds__(256) void eigval_kernel(
    const float* __restrict__ dg, const float* __restrict__ od,
    float* __restrict__ L, int n) {
  const int t = blockIdx.x * blockDim.x + threadIdx.x;
  if (t >= n) return;
  float lo = 3.4e38f, hi = -3.4e38f;
  for (int i = 0; i < n; ++i) {
    float r = 0.0f;
    if (i > 0) r += fabsf(od[i - 1]);
    if (i < n - 1) r += fabsf(od[i]);
    float d = dg[i];
    lo = fminf(lo, d - r);
    hi = fmaxf(hi, d + r);
  }
  for (int it = 0; it < 44; ++it) {
    float mid = 0.5f * (lo + hi);
    if (sturm_count(dg, od, n, mid) >= t + 1) hi = mid;
    else lo = mid;
  }
  L[t] = 0.5f * (lo + hi);
}

// ---------------------------------------------------------------------------
// Eigenvector columns via the tridiagonal three-term recurrence.
// ---------------------------------------------------------------------------
__global__ __launch_bounds__(256) void eigvec_kernel(
    const float* __restrict__ dg, const float* __restrict__ od,
    const float* __restrict__ L, float* __restrict__ Q, int n) {
  const int t = blockIdx.x * blockDim.x + threadIdx.x;
  if (t >= n) return;
  const float lam = L[t];
  float vkm1 = 1.0f;
  float e0 = od[0];
  if (fabsf(e0) < 1e-20f) e0 = (e0 < 0.0f) ? -1e-20f : 1e-20f;
  float vk = (lam - dg[0]) / e0;
  Q[(size_t)0 * n + t] = vkm1;
  Q[(size_t)1 * n + t] = vk;
  float sumsq = vkm1 * vkm1 + vk * vk;
  for (int k = 1; k < n - 1; ++k) {
    float ek = od[k];
    if (fabsf(ek) < 1e-20f) ek = (ek < 0.0f) ? -1e-20f : 1e-20f;
    float vkp1 = ((lam - dg[k]) * vk - od[k - 1] * vkm1) / ek;
    Q[(size_t)(k + 1) * n + t] = vkp1;
    sumsq += vkp1 * vkp1;
    vkm1 = vk;
    vk = vkp1;
    if (fabsf(vk) > 1e18f) {  // crude overflow guard on running state
      vk *= 1e-18f;
      vkm1 *= 1e-18f;
      sumsq *= 1e-36f;
    }
  }
  const float rn = rsqrtf(fmaxf(sumsq, 1e-30f));
  for (int k = 0; k < n; ++k) Q[(size_t)k * n + t] *= rn;
}

#if HAVE_TDM
// ---------------------------------------------------------------------------
// Tensor Data Mover: 2D tile (tileW x tileH f32 elements, row stride
// `strideElems`) from global -> LDS at byte offset `ldsOff`, with the TDM
// pad feature inserting `padAmountEnc+1` DWORDs after every 2^(padIntEnc+1)
// DWORDs (used to realize the bank-conflict-free LDS pitches).
// D# layout per CDNA5 ISA ch. 8 (group0 128b, group1 256b; trailing groups
// zero for 2D tiles). workgroup_mask=0 (not in a cluster).
// This toolchain's builtin arity is 6: (v4u, v8i, v4i, v4i, v8i, i32 cpol).
// ---------------------------------------------------------------------------
__device__ __forceinline__ void tdm_load_2d(unsigned ldsOff,
                                            const float* gptr,
                                            unsigned strideElems,
                                            unsigned tileW, unsigned tileH,
                                            unsigned padIntEnc,
                                            unsigned padAmtEnc) {
  const unsigned long long ga = (unsigned long long)(const void*)gptr;
  const unsigned tdim1 = 1u << 20;  // generous dim1 extent (OOB guard only)
  v4u g0;
  g0.x = 1u;  // count=1, is_restore=0, gather_mode=0
  g0.y = ldsOff;
  g0.z = (unsigned)(ga & 0xFFFFFFFFu);
  g0.w = (unsigned)((ga >> 32) & 0x1FFFFFFu) | (2u << 30);  // type=2 (image)
  v8i g1;
  g1[0] = (int)((2u << 16)                  // data_size: 4 bytes
                | (1u << 20)                // pad_enable
                | (padIntEnc << 22) | (padAmtEnc << 25));
  g1[1] = (int)((strideElems & 0xFFFFu) << 16);        // tensor_dim0[15:0]
  g1[2] = (int)(((strideElems >> 16) & 0xFFFFu)        // tensor_dim0[31:16]
                | ((tdim1 & 0xFFFFu) << 16));          // tensor_dim1[15:0]
  g1[3] = (int)(((tdim1 >> 16) & 0xFFFFu)              // tensor_dim1[31:16]
                | (tileW << 16));                      // tile_dim0
  g1[4] = (int)tileH;                                  // tile_dim1 (dim2=0)
  g1[5] = (int)strideElems;                            // dim0_stride[31:0]
  g1[6] = 0;                                           // stride hi / dim1_stride
  g1[7] = 0;
  v4i z4;
  z4[0] = z4[1] = z4[2] = z4[3] = 0;
  v8i z8;
  z8[0] = z8[1] = z8[2] = z8[3] = 0;
  z8[4] = z8[5] = z8[6] = z8[7] = 0;
  __builtin_amdgcn_tensor_load_to_lds(g0, g1, z4, z4, z8, 0);
}
#endif

// ---------------------------------------------------------------------------
// Fused WMMA f32 GEMM: C[rows x 64] = op(A) @ Bmat  (+ fused epilogue)
//   TRANS_A=false : Aeff(r,k) = Amat[r*lda + k]
//   TRANS_A=true  : Aeff(r,k) = Amat[k*lda + r]
// 128 threads = 4 waves, 64x64 tile, K_BLK=32 through LDS.
// B panel transposed to BsT[col][k] (pitch 36) so each WMMA B-fragment is one
// contiguous 8B ds_load_b64, bank-conflict-free. A panel copied by the TDM
// (padded row-major), fragments either contiguous b64 (notrans) or a
// 2addr-b32 pair (trans).
// MODE epilogues:
//   0: C = acc - E0[idx]                       (R = A@Z - Y)
//   1: C = -(acc + E2[0]*(E0[idx]-E1[idx]))    (D = -(AtR + kappa(Z-U)))
//   2: C = max(E0[row], E2[0]) * acc           (W = Lc .* (Qt D))
//   3: C = E0[idx] + E1[row]*acc               (OUT = Z + eta.*(Q W))
// ---------------------------------------------------------------------------
template <int MODE, bool TRANS_A>
__global__ __launch_bounds__(128) void gemm_wmma(
    const float* __restrict__ Amat, int lda,
    const float* __restrict__ Bmat, int KD,
    float* __restrict__ Cmat,
    const float* __restrict__ E0, const float* __restrict__ E1,
    const float* __restrict__ E2) {
  constexpr int K_BLK = 32;
  __shared__ float BsT[64][36];  // [col][k], pitch 36 (conflict-free, 16B rows)
  constexpr int AR = TRANS_A ? K_BLK : 64;
  constexpr int AC = TRANS_A ? 72 : 36;  // trans: [k][row] pitch 72
  __shared__ float As[AR][AC];           // notrans: [row][k] pitch 36

  const int tid = threadIdx.x;  // 0..127
  const int lane = tid & 31;
  const int wave = tid >> 5;    // 0..3 -> M-subtile
  const int rowBase = blockIdx.x * 64;
  const int ml = lane & 15;            // M/N lane index
  const int khalf = (lane >> 4) << 1;  // 0 or 2: K offset of this half-wave

  v8f acc[4];
#pragma unroll
  for (int j = 0; j < 4; ++j)
#pragma unroll
    for (int i = 0; i < 8; ++i) acc[j][i] = 0.0f;

#if HAVE_TDM
  const unsigned asOff = (unsigned)(size_t)&As[0][0];
#endif

  for (int k0 = 0; k0 < KD; k0 += K_BLK) {
    // ---- A panel: TDM async DMA (one instruction, issued by wave 0) ----
#if HAVE_TDM
    if (wave == 0) {
      if constexpr (TRANS_A) {
        // tile [K_BLK rows][64 cols] -> As[k][row], pitch 64->72 (pad 8/64)
        tdm_load_2d(asOff, Amat + (size_t)k0 * lda + rowBase, (unsigned)lda,
                    64u, (unsigned)K_BLK, 5u, 7u);
      } else {
        // tile [64 rows][K_BLK cols] -> As[row][k], pitch 32->36 (pad 4/32)
        tdm_load_2d(asOff, Amat + (size_t)rowBase * lda + k0, (unsigned)lda,
                    (unsigned)K_BLK, 64u, 4u, 3u);
      }
    }
#else
    if constexpr (TRANS_A) {
      for (int s = tid; s < (K_BLK * 64) / 4; s += 128) {
        const int r = s >> 4;
        const int c4 = (s & 15) * 4;
        *(float4*)&As[r][c4] =
            *(const float4*)&Amat[(size_t)(k0 + r) * lda + rowBase + c4];
      }
    } else {
      for (int s = tid; s < (64 * K_BLK) / 4; s += 128) {
        const int r = s >> 3;
        const int c4 = (s & 7) * 4;
        *(float4*)&As[r][c4] =
            *(const float4*)&Amat[(size_t)(rowBase + r) * lda + k0 + c4];
      }
    }
#endif

    // ---- B panel: transpose-stage [K_BLK][64] -> BsT[col][k] ----
    // lanes write consecutive k at fixed col -> conflict-free b32 stores
#pragma unroll
    for (int s = tid; s < (K_BLK * 64) / 4; s += 128) {
      const int r = s & 31;          // k within panel
      const int c4 = (s >> 5) * 4;   // starting column
      const float4 v = *(const float4*)&Bmat[(size_t)(k0 + r) * 64 + c4];
      BsT[c4 + 0][r] = v.x;
      BsT[c4 + 1][r] = v.y;
      BsT[c4 + 2][r] = v.z;
      BsT[c4 + 3][r] = v.w;
    }
    // prefetch next K panel of B (global_prefetch_b8)
    if (k0 + K_BLK < KD)
      __builtin_prefetch(&Bmat[(size_t)(k0 + K_BLK) * 64 + tid * 4], 0, 0);

#if HAVE_TDM
    if (wave == 0) __builtin_amdgcn_s_wait_tensorcnt(0);
#endif
    __syncthreads();

    // ---- compute: 8 k-steps x 4 N-subtiles of v_wmma_f32_16x16x4_f32 ----
#pragma unroll
    for (int kk = 0; kk < K_BLK; kk += 4) {
      const int ko = kk + khalf;
      v2f a;
      if constexpr (TRANS_A) {
        a.x = As[ko][wave * 16 + ml];
        a.y = As[ko + 1][wave * 16 + ml];
      } else {
        a = *(const v2f*)&As[wave * 16 + ml][ko];  // contiguous b64
      }
#pragma unroll
      for (int j = 0; j < 4; ++j) {
        const v2f b = *(const v2f*)&BsT[j * 16 + ml][ko];  // contiguous b64
        acc[j] = __builtin_amdgcn_wmma_f32_16x16x4_f32(
            false, a, false, b, (short)0, acc[j], false, false);
      }
    }
    __syncthreads();
  }

  // ---- fused epilogue ----
  float s0 = 0.0f;
  if constexpr (MODE == 1 || MODE == 2) s0 = E2[0];
  const int rhalf = (lane >> 4) * 8;
  const int grow0 = rowBase + wave * 16;
#pragma unroll
  for (int j = 0; j < 4; ++j) {
    const int col = j * 16 + ml;
#pragma unroll
    for (int i = 0; i < 8; ++i) {
      const int grow = grow0 + rhalf + i;
      const size_t idx = (size_t)grow * 64 + col;
      const float v = acc[j][i];
      float o;
      if constexpr (MODE == 0) {
        o = v - E0[idx];
      } else if constexpr (MODE == 1) {
        o = -(v + s0 * (E0[idx] - E1[idx]));
      } else if constexpr (MODE == 2) {
        o = fmaxf(E0[grow], s0) * v;
      } else {
        o = E0[idx] + E1[grow] * v;
      }
      Cmat[idx] = o;
    }
  }
}

// ---------------------------------------------------------------------------
// Launch. d_ws layout (floats): Q[N*N] | L[N] | R[M*B] | D[N*B] | W[N*B]
// (~70 MB total, assumed <= ws_size).
// ---------------------------------------------------------------------------
extern "C" void kernel_launch(void* const* d_in, const int* in_sizes, int n_in,
                              void* d_out, int out_size, void* d_ws,
                              size_t ws_size, hipStream_t stream) {
  const float* z = (const float*)d_in[0];
  const float* u = (const float*)d_in[1];
  const float* y = (const float*)d_in[2];
  const float* A = (const float*)d_in[3];
  const float* kappa = (const float*)d_in[4];
  const float* eps = (const float*)d_in[5];
  const float* eta = (const float*)d_in[6];
  const float* dg = (const float*)d_in[7];
  const float* od = (const float*)d_in[8];
  float* out = (float*)d_out;

  float* ws = (float*)d_ws;
  float* Q = ws;
  ws += (size_t)N_ * N_;
  float* Lv = ws;
  ws += N_;
  float* Rb = ws;
  ws += (size_t)M_ * B_;
  float* Db = ws;
  ws += (size_t)N_ * B_;
  float* Wb = ws;

  // 1) tridiagonal eigendecomposition
  eigval_kernel<<<N_ / 256, 256, 0, stream>>>(dg, od, Lv, N_);
  eigvec_kernel<<<N_ / 256, 256, 0, stream>>>(dg, od, Lv, Q, N_);

  // 2) R = A @ Z - Y                       [M,B]
  gemm_wmma<0, false><<<M_ / 64, 128, 0, stream>>>(A, N_, z, N_, Rb, y,
                                                   nullptr, nullptr);
  // 3) D = -(A^T @ R + kappa*(Z-U))        [N,B]
  gemm_wmma<1, true><<<N_ / 64, 128, 0, stream>>>(A, N_, Rb, M_, Db, z, u,
                                                  kappa);
  // 4) W = max(L,eps) .* (Q^T @ D)         [N,B]
  gemm_wmma<2, true><<<N_ / 64, 128, 0, stream>>>(Q, N_, Db, N_, Wb, Lv,
                                                  nullptr, eps);
  // 5) OUT = Z + eta .* (Q @ W)            [N,B]
  gemm_wmma<3, false><<<N_ / 64, 128, 0, stream>>>(Q, N_, Wb, N_, out, z, eta,
                                                   nullptr);
}